// DGPool_36807869727433
// MI455X (gfx1250) — compile-verified
//
#include <hip/hip_runtime.h>
#include <math.h>

#define NUM_NODES 10000
#define KSEL      5000
#define DDIM      512
#define BB        32
#define EPSF      1e-8f
#define CHUNK     128

typedef __attribute__((ext_vector_type(2))) float v2f;
typedef __attribute__((ext_vector_type(8))) float v8f;

// ---------------- kernel 0: zero outputs + per-batch accumulators ----------
__global__ void k_init(float* __restrict__ out, float* __restrict__ bsum,
                       float* __restrict__ bsumsq) {
    int i = blockIdx.x * 256 + threadIdx.x;
    if (i < BB * DDIM + 1) out[i] = 0.0f;
    if (i < BB) { bsum[i] = 0.0f; bsumsq[i] = 0.0f; }
}

// ---------------- kernel 1: wn = v / (||v|| + eps) -------------------------
__global__ void k_normw(const float* __restrict__ v, float* __restrict__ wn) {
    __shared__ float red[256];
    int t = threadIdx.x;
    float acc = 0.0f;
    for (int i = t; i < DDIM; i += 256) { float x = v[i]; acc += x * x; }
    red[t] = acc; __syncthreads();
    for (int s = 128; s > 0; s >>= 1) { if (t < s) red[t] += red[t + s]; __syncthreads(); }
    float inv = 1.0f / (sqrtf(red[0]) + EPSF);
    for (int i = t; i < DDIM; i += 256) wn[i] = v[i] * inv;
}

// ---------------- kernel 2: scores via V_WMMA_F32_16X16X4_F32 --------------
// One wave = one 16-row tile. B-fragment carries wn replicated across all 16
// columns, so D[m, n] == score(row m) for every n regardless of B's column
// mapping. K <-> (vgpr, lane-half) mapping mirrors the documented A layout:
// K = vgpr_idx + 2*(lane>=16).
__global__ void __launch_bounds__(256)
k_scores(const float* __restrict__ x, const float* __restrict__ wn,
         float* __restrict__ scores, float* __restrict__ bsum,
         float* __restrict__ bsumsq) {
    const int lane  = threadIdx.x & 31;
    const int wave  = threadIdx.x >> 5;
    const long tile = (long)blockIdx.x * 8 + wave;     // 16-row tile id
    const long row_base = tile * 16;
    const int  m     = lane & 15;
    const int  khalf = (lane >> 4) << 1;               // 0 or 2
    const float* __restrict__ pa = x + (row_base + m) * DDIM + khalf;
    const float* __restrict__ pb = wn + khalf;

    v8f c0 = {}; v8f c1 = {};
    for (int kc0 = 0; kc0 < DDIM; kc0 += 32) {
        __builtin_prefetch(pa + kc0 + 64, 0, 3);       // global_prefetch_b8
#pragma unroll
        for (int kc = kc0; kc < kc0 + 32; kc += 8) {
            v2f a0 = *(const v2f*)(pa + kc);
            v2f b0 = *(const v2f*)(pb + kc);
            v2f a1 = *(const v2f*)(pa + kc + 4);
            v2f b1 = *(const v2f*)(pb + kc + 4);
            c0 = __builtin_amdgcn_wmma_f32_16x16x4_f32(false, a0, false, b0,
                                                       (short)0, c0, false, false);
            c1 = __builtin_amdgcn_wmma_f32_16x16x4_f32(false, a1, false, b1,
                                                       (short)0, c1, false, false);
        }
    }
    v8f c = c0 + c1;

    // C/D layout: vgpr r -> row r (lanes 0-15) / row r+8 (lanes 16-31)
    if ((lane & 15) == 0) {
        const int roff = (lane >> 4) * 8;
        float s1 = 0.0f, s2 = 0.0f;
#pragma unroll
        for (int r = 0; r < 8; ++r) {
            float sc = c[r];
            scores[row_base + roff + r] = sc;
            s1 += sc; s2 += sc * sc;
        }
        int b = (int)(row_base / NUM_NODES);           // tiles never straddle batches
        atomicAdd(&bsum[b], s1);
        atomicAdd(&bsumsq[b], s2);
    }
}

// ---------------- kernel 3: per-batch sigmoid + radix-select + loss --------
__global__ void __launch_bounds__(256)
k_select(const float* __restrict__ scores, float* __restrict__ s_glob,
         const float* __restrict__ bsum, const float* __restrict__ bsumsq,
         int* __restrict__ sel, float* __restrict__ out_loss) {
    __shared__ float    s_lds[NUM_NODES];              // 40 KB of the 320 KB pool
    __shared__ unsigned hist[256];
    __shared__ unsigned sh_u[3];                       // prefix, kth, tie-count
    __shared__ float    sh_loss;
    __shared__ unsigned selCnt, tieCnt;

    const int b = blockIdx.x;
    const int t = threadIdx.x;

    const float mu  = bsum[b] * (1.0f / NUM_NODES);
    float var = bsumsq[b] * (1.0f / NUM_NODES) - mu * mu;
    var = var > 0.0f ? var : 0.0f;
    const float inv = 1.0f / (sqrtf(var) + EPSF);

    const float* __restrict__ sc = scores + (long)b * NUM_NODES;
    for (int n = t; n < NUM_NODES; n += 256) {
        float z = (sc[n] - mu) * inv;
        float s = 1.0f / (1.0f + __expf(-z));
        s_lds[n] = s;
        s_glob[(long)b * NUM_NODES + n] = s;
    }
    if (t == 0) { sh_u[0] = 0u; sh_u[1] = KSEL; sh_loss = 0.0f; selCnt = 0u; tieCnt = 0u; }
    __syncthreads();

    // 4-pass radix select for the KSEL-th largest sigmoid (positive floats:
    // uint bit pattern is order-preserving).
    unsigned mask = 0u;
    for (int pass = 0; pass < 4; ++pass) {
        const int shift = 24 - 8 * pass;
        hist[t] = 0u; __syncthreads();
        const unsigned prefix = sh_u[0];
        for (int n = t; n < NUM_NODES; n += 256) {
            unsigned key = __float_as_uint(s_lds[n]);
            if ((key & mask) == prefix) atomicAdd(&hist[(key >> shift) & 255u], 1u);
        }
        __syncthreads();
        if (t == 0) {
            unsigned kth = sh_u[1];
            unsigned cum = 0u; int bin = 255;
            for (; bin > 0; --bin) {
                unsigned h = hist[bin];
                if (cum + h >= kth) break;
                cum += h;
            }
            sh_u[0] = prefix | ((unsigned)bin << shift);
            sh_u[1] = kth - cum;
            sh_u[2] = hist[bin];                       // tie count after last pass
        }
        __syncthreads();
        mask |= 0xFFu << shift;
    }
    const unsigned thresh = sh_u[0];
    const unsigned mtake  = sh_u[1];                   // #ties included in top-K
    const unsigned tcount = sh_u[2];
    const float    sTh    = __uint_as_float(thresh);

    // loss: strict sides elementwise, ties via closed form (equal logs)
    float acc = 0.0f;
    for (int n = t; n < NUM_NODES; n += 256) {
        float s = s_lds[n];
        unsigned key = __float_as_uint(s);
        if (key > thresh)      acc += __logf(s + EPSF);
        else if (key < thresh) acc += __logf(1.0f - s + EPSF);
    }
    atomicAdd(&sh_loss, acc);
    __syncthreads();
    if (t == 0) {
        float l = sh_loss + (float)mtake * __logf(sTh + EPSF)
                          + (float)(tcount - mtake) * __logf(1.0f - sTh + EPSF);
        float loss_b = -l * (1.0f / NUM_NODES);
        atomicAdd(out_loss, loss_b * (1.0f / BB));     // mean over batches
    }

    // compact exactly KSEL selected node indices
    for (int n = t; n < NUM_NODES; n += 256) {
        unsigned key = __float_as_uint(s_lds[n]);
        if (key > thresh) {
            unsigned pos = atomicAdd(&selCnt, 1u);
            sel[(long)b * KSEL + pos] = n;
        } else if (key == thresh) {
            unsigned tp = atomicAdd(&tieCnt, 1u);
            if (tp < mtake) {
                unsigned pos = atomicAdd(&selCnt, 1u);
                sel[(long)b * KSEL + pos] = n;
            }
        }
    }
}

// ---------------- kernel 4: pooled = mean of sig*x over selected rows ------
__global__ void __launch_bounds__(256)
k_pool(const float* __restrict__ x, const float* __restrict__ s_glob,
       const int* __restrict__ sel, float* __restrict__ out) {
    const int nchunks = (KSEL + CHUNK - 1) / CHUNK;    // 40
    const int b     = blockIdx.x / nchunks;
    const int chunk = blockIdx.x % nchunks;
    const int t  = threadIdx.x;
    const int d  = t * 2;                              // 256 threads * float2 = 2KB row
    const int r0 = chunk * CHUNK;
    const int r1 = (r0 + CHUNK < KSEL) ? (r0 + CHUNK) : KSEL;

    float ax = 0.0f, ay = 0.0f;
    for (int r = r0; r < r1; ++r) {
        int   n   = sel[(long)b * KSEL + r];
        float wgt = s_glob[(long)b * NUM_NODES + n];
        const float2 xv = *(const float2*)(x + ((long)b * NUM_NODES + n) * DDIM + d);
        ax += wgt * xv.x; ay += wgt * xv.y;
    }
    const float invK = 1.0f / KSEL;
    atomicAdd(&out[b * DDIM + d],     ax * invK);
    atomicAdd(&out[b * DDIM + d + 1], ay * invK);
}

// ---------------- launch ---------------------------------------------------
extern "C" void kernel_launch(void* const* d_in, const int* in_sizes, int n_in,
                              void* d_out, int out_size, void* d_ws, size_t ws_size,
                              hipStream_t stream) {
    const float* x = (const float*)d_in[0];            // (B*N, D) fp32
    const float* v = (const float*)d_in[1];            // (D, 1)  fp32
    float* out = (float*)d_out;                        // [B*D pooled | 1 loss]

    // workspace layout (floats, then ints)
    float* ws      = (float*)d_ws;
    float* wn      = ws;                               // 512
    float* scores  = wn + DDIM;                        // B*N
    float* s_glob  = scores + (long)BB * NUM_NODES;    // B*N
    float* bsum    = s_glob + (long)BB * NUM_NODES;    // B
    float* bsumsq  = bsum + BB;                        // B
    int*   sel     = (int*)(bsumsq + BB);              // B*KSEL ints

    k_init <<<(BB * DDIM + 1 + 255) / 256, 256, 0, stream>>>(out, bsum, bsumsq);
    k_normw<<<1, 256, 0, stream>>>(v, wn);

    const int tiles = BB * NUM_NODES / 16;             // 20000 (N % 16 == 0)
    k_scores<<<tiles / 8, 256, 0, stream>>>(x, wn, scores, bsum, bsumsq);

    k_select<<<BB, 256, 0, stream>>>(scores, s_glob, bsum, bsumsq, sel,
                                     out + BB * DDIM);

    k_pool  <<<BB * ((KSEL + CHUNK - 1) / CHUNK), 256, 0, stream>>>(x, s_glob, sel, out);
}